// BrainGNNEncoder_11312943857689
// MI455X (gfx1250) — compile-verified
//
#include <hip/hip_runtime.h>
#include <hip/hip_bf16.h>

#define BN_EPS 1e-5f

typedef __attribute__((ext_vector_type(2))) float v2f;
typedef __attribute__((ext_vector_type(8))) float v8f;

// ---------------- degree / normalization ----------------
__global__ void k_init_deg(float* deg, int N) {
  int i = blockIdx.x * blockDim.x + threadIdx.x;
  if (i < N) deg[i] = 1.0f; // self-loop contributes weight 1
}

__global__ void k_acc_deg(const long long* __restrict__ col,
                          const float* __restrict__ ew,
                          float* deg, int E) {
  int e = blockIdx.x * blockDim.x + threadIdx.x;
  if (e < E) atomicAdd(&deg[(int)col[e]], fabsf(ew[e]));
}

__global__ void k_dinv(float* deg, int N) {
  int i = blockIdx.x * blockDim.x + threadIdx.x;
  if (i < N) deg[i] = rsqrtf(deg[i]); // deg >= 1 always (self loop)
}

// ---------------- GEMM: Y[n_rows,64] = X[n_rows,K] @ W[K,64] -------------
// fp32 WMMA 16x16x4. One wave computes a 16-row strip x 64 cols (4 tiles).
// K must be a multiple of 4 (100 and 64 both are); n_rows multiple of 16.
// W is staged in LDS K-pair-interleaved: sW2[(k>>1)*128 + c*2 + (k&1)]
// so each B fragment {W[k][c], W[k+1][c]} is one aligned ds_load_b64.
__global__ __launch_bounds__(256) void k_gemm_wmma(
    const float* __restrict__ X, const float* __restrict__ W,
    float* __restrict__ Y, int n_rows, int K) {
  __shared__ float sW2[(100 / 2) * 128]; // max K = 100 -> 25.6 KB LDS
  int tid = threadIdx.x;
  int kw = K * 64;
  for (int i = tid; i < kw; i += 256) {
    int k = i >> 6, c = i & 63;
    sW2[(k >> 1) * 128 + c * 2 + (k & 1)] = W[i];
  }
  __syncthreads();

  int wave = tid >> 5;
  int lane = tid & 31;
  int strip = blockIdx.x * 8 + wave; // 16-row strip index (wave-uniform)
  if (strip * 16 >= n_rows) return;  // whole-wave exit: EXEC stays all-1s

  int row0 = strip * 16;
  int m = lane & 15;            // row within strip (A) / column within tile (B)
  int koff = (lane >> 4) * 2;   // lanes 0-15 hold K+0,K+1 ; lanes 16-31 hold K+2,K+3

  v8f acc0 = {}, acc1 = {}, acc2 = {}, acc3 = {};
  const float* xrow = X + (size_t)(row0 + m) * K;

  for (int k = 0; k < K; k += 4) {
    v2f a = *(const v2f*)(xrow + k + koff);      // global_load_b64
    const v2f* wp = (const v2f*)&sW2[((k + koff) >> 1) * 128];
    v2f b0 = wp[m];                               // ds_load_b64 each
    v2f b1 = wp[m + 16];
    v2f b2 = wp[m + 32];
    v2f b3 = wp[m + 48];
    acc0 = __builtin_amdgcn_wmma_f32_16x16x4_f32(false, a, false, b0, (short)0, acc0, false, false);
    acc1 = __builtin_amdgcn_wmma_f32_16x16x4_f32(false, a, false, b1, (short)0, acc1, false, false);
    acc2 = __builtin_amdgcn_wmma_f32_16x16x4_f32(false, a, false, b2, (short)0, acc2, false, false);
    acc3 = __builtin_amdgcn_wmma_f32_16x16x4_f32(false, a, false, b3, (short)0, acc3, false, false);
  }

  // C/D layout: VGPR r -> row r (lanes 0-15) / row r+8 (lanes 16-31), col = lane%16
  int r_add = (lane >> 4) * 8;
#pragma unroll
  for (int r = 0; r < 8; ++r) {
    float* yr = Y + (size_t)(row0 + r + r_add) * 64 + m;
    yr[0]  = acc0[r];
    yr[16] = acc1[r];
    yr[32] = acc2[r];
    yr[48] = acc3[r];
  }
}

// ---------------- aggregation ----------------
// self-loop term initializes agg: agg[i][:] = dinv[i]^2 * xw[i][:]
__global__ void k_selfloop(const float* __restrict__ xw,
                           const float* __restrict__ dinv,
                           float* __restrict__ agg, int N) {
  int t = blockIdx.x * blockDim.x + threadIdx.x;
  if (t >= N * 64) return;
  int i = t >> 6;
  float d = dinv[i];
  agg[t] = xw[t] * d * d;
}

// one thread per (edge, 4-feature group): agg[col] += norm * xw[row]
__global__ void k_agg_edges(const long long* __restrict__ row,
                            const long long* __restrict__ col,
                            const float* __restrict__ ew,
                            const float* __restrict__ dinv,
                            const float* __restrict__ xw,
                            float* __restrict__ agg, int E) {
  int t = blockIdx.x * blockDim.x + threadIdx.x;
  int e = t >> 4;
  if (e >= E) return;
  int c0 = (t & 15) * 4;
  int r = (int)row[e];
  int c = (int)col[e];
  float norm = dinv[r] * fabsf(ew[e]) * dinv[c];
  const float4 v = *(const float4*)(xw + (size_t)r * 64 + c0);
  float* dst = agg + (size_t)c * 64 + c0;
  atomicAdd(dst + 0, norm * v.x);
  atomicAdd(dst + 1, norm * v.y);
  atomicAdd(dst + 2, norm * v.z);
  atomicAdd(dst + 3, norm * v.w);
}

// bias + BatchNorm(eval) + ReLU, in place
__global__ void k_post(float* __restrict__ agg,
                       const float* __restrict__ bias,
                       const float* __restrict__ gamma,
                       const float* __restrict__ beta,
                       const float* __restrict__ rmean,
                       const float* __restrict__ rvar, int N) {
  int t = blockIdx.x * blockDim.x + threadIdx.x;
  if (t >= N * 64) return;
  int c = t & 63;
  float v = agg[t] + bias[c];
  v = (v - rmean[c]) * rsqrtf(rvar[c] + BN_EPS) * gamma[c] + beta[c];
  agg[t] = fmaxf(v, 0.0f);
}

// ---------------- pooling ----------------
__global__ void k_zero_pool(float* out, float* cnt, int G) {
  int i = blockIdx.x * blockDim.x + threadIdx.x;
  if (i < G * 128) out[i] = 0.0f;
  if (i < G) cnt[i] = 0.0f;
}

__global__ void k_pool(const float* __restrict__ act,
                       const long long* __restrict__ batch,
                       float* __restrict__ out, float* __restrict__ cnt, int N) {
  int t = blockIdx.x * blockDim.x + threadIdx.x;
  if (t >= N * 64) return;
  int i = t >> 6, c = t & 63;
  int g = (int)batch[i];
  atomicAdd(&out[g * 128 + 64 + c], act[(size_t)i * 64 + c]); // sum half
  if (c == 0) atomicAdd(&cnt[g], 1.0f);
}

__global__ void k_pool_mean(float* __restrict__ out, const float* __restrict__ cnt, int G) {
  int t = blockIdx.x * blockDim.x + threadIdx.x;
  if (t >= G * 64) return;
  int g = t >> 6, c = t & 63;
  out[g * 128 + c] = out[g * 128 + 64 + c] / fmaxf(cnt[g], 1.0f);
}

// ---------------- driver ----------------
extern "C" void kernel_launch(void* const* d_in, const int* in_sizes, int n_in,
                              void* d_out, int out_size, void* d_ws, size_t ws_size,
                              hipStream_t stream) {
  (void)n_in; (void)ws_size;
  const float*     x     = (const float*)d_in[0];
  const long long* ei    = (const long long*)d_in[1];
  const float*     ew    = (const float*)d_in[2];
  const long long* batch = (const long long*)d_in[3];

  const int N   = in_sizes[3];          // 100000
  const int E   = in_sizes[2];          // 1600000
  const int CIN = in_sizes[0] / N;      // 100
  const int G   = out_size / 128;       // 1000
  float* out = (float*)d_out;

  // workspace carve-up (floats): deg/dinv[N] | xw[N*64] | act[N*64] | cnt[G]
  float* ws  = (float*)d_ws;
  float* deg = ws;
  float* xw  = deg + N;
  float* act = xw + (size_t)N * 64;
  float* cnt = act + (size_t)N * 64;

  const long long* row = ei;       // edge_index[0]
  const long long* col = ei + E;   // edge_index[1]

  const int B = 256;
  k_init_deg<<<(N + B - 1) / B, B, 0, stream>>>(deg, N);
  k_acc_deg<<<(E + B - 1) / B, B, 0, stream>>>(col, ew, deg, E);
  k_dinv<<<(N + B - 1) / B, B, 0, stream>>>(deg, N);

  int strips      = N / 16;                 // 6250
  int gemm_blocks = (strips + 7) / 8;       // 8 waves per block
  int nf_blocks   = (N * 64 + B - 1) / B;   // 6.4M threads
  int ef_blocks   = (E * 16 + B - 1) / B;   // 25.6M threads

  const float* src = x;
  for (int l = 0; l < 3; ++l) {
    const float* W  = (const float*)d_in[4 + 6 * l + 0];
    const float* bb = (const float*)d_in[4 + 6 * l + 1];
    const float* gg = (const float*)d_in[4 + 6 * l + 2];
    const float* bt = (const float*)d_in[4 + 6 * l + 3];
    const float* rm = (const float*)d_in[4 + 6 * l + 4];
    const float* rv = (const float*)d_in[4 + 6 * l + 5];
    int K = (l == 0) ? CIN : 64;

    k_gemm_wmma<<<gemm_blocks, 256, 0, stream>>>(src, W, xw, N, K);
    k_selfloop<<<nf_blocks, B, 0, stream>>>(xw, deg, act, N);
    k_agg_edges<<<ef_blocks, B, 0, stream>>>(row, col, ew, deg, xw, act, E);
    k_post<<<nf_blocks, B, 0, stream>>>(act, bb, gg, bt, rm, rv, N);
    src = act;
  }

  k_zero_pool<<<(G * 128 + B - 1) / B, B, 0, stream>>>(out, cnt, G);
  k_pool<<<nf_blocks, B, 0, stream>>>(act, batch, out, cnt, N);
  k_pool_mean<<<(G * 64 + B - 1) / B, B, 0, stream>>>(out, cnt, G);
}